// PortfolioOptimization_40175124087521
// MI455X (gfx1250) — compile-verified
//
#include <hip/hip_runtime.h>
#include <math.h>

// ---------------------------------------------------------------------------
// Batched QP via projected gradient descent (portfolio optimization).
//   rets:        (B, 128)  fp32
//   covmat_sqrt: (B, 128, 128) fp32
//   gamma:       (B,) fp32
//   out w:       (B, 128) fp32
// One workgroup (256 threads = 8 wave32) per sample.
//   Phase 0: A staged global->LDS by the Tensor Data Mover (one
//            tensor_load_to_lds DMA, padded rows), fallback to b128 loads.
//   Phase 1: Q = gamma^2 * A^T A via V_WMMA_F32_16X16X4_F32, Q kept in LDS.
//   Phase 2: 500 PGD iterations, fully LDS-resident (matvec + capped-simplex
//            projection with redundant per-wave bisection, 2 barriers/iter).
// Roofline: 134 MB read once (~6 us @ 23.3 TB/s) => compute/latency bound on
// the serial PGD chain, so everything lives in LDS (A 64KB + Q 64KB < 320KB).
// ---------------------------------------------------------------------------

typedef __attribute__((ext_vector_type(2))) float v2f;
typedef __attribute__((ext_vector_type(8))) float v8f;
typedef __attribute__((ext_vector_type(4))) unsigned int u32x4;
typedef __attribute__((ext_vector_type(8))) int i32x8;
typedef __attribute__((ext_vector_type(4))) int i32x4;

#define N_ASSETS   128
#define LSTRIDE    132          // padded LDS row stride (floats); TDM pad: 128 DW + 4 DW
#define PGD_ITERS  500
#define BIS_ITERS  60
#define MW         1.0f

#if defined(__gfx1250__) && __has_builtin(__builtin_amdgcn_tensor_load_to_lds)
#define USE_TDM 1
#else
#define USE_TDM 0
#endif

__device__ __forceinline__ float wred_sum(float x) {
#pragma unroll
  for (int off = 16; off > 0; off >>= 1) x += __shfl_xor(x, off, 32);
  return x;
}
__device__ __forceinline__ float wred_min(float x) {
#pragma unroll
  for (int off = 16; off > 0; off >>= 1) x = fminf(x, __shfl_xor(x, off, 32));
  return x;
}
__device__ __forceinline__ float wred_max(float x) {
#pragma unroll
  for (int off = 16; off > 0; off >>= 1) x = fmaxf(x, __shfl_xor(x, off, 32));
  return x;
}
__device__ __forceinline__ float clip01(float x) {
  return fminf(fmaxf(x, 0.0f), MW);
}

__global__ __launch_bounds__(256)
void PortfolioPGD_kernel(const float* __restrict__ rets,
                         const float* __restrict__ covmat_sqrt,
                         const float* __restrict__ gamma,
                         float* __restrict__ out) {
  extern __shared__ float smem[];
  float* A_s    = smem;                    // 128 * 132
  float* Q_s    = A_s + N_ASSETS * LSTRIDE;// 128 * 132
  float* rets_s = Q_s + N_ASSETS * LSTRIDE;// 128
  float* w_s    = rets_s + N_ASSETS;       // 128
  float* v_s    = w_s + N_ASSETS;          // 128
  float* red_s  = v_s + N_ASSETS;          // 8
  float* step_s = red_s + 8;               // 1

  const int b    = blockIdx.x;
  const int tid  = threadIdx.x;
  const int lane = tid & 31;
  const int wave = tid >> 5;

  const float* A_g = covmat_sqrt + (size_t)b * N_ASSETS * N_ASSETS;

  // ------------------------------------------------------------------
  // Phase 0: stage A (64 KB) into LDS with row padding 128->132 floats.
  // ------------------------------------------------------------------
#if USE_TDM
  if (wave == 0) {
    // Tensor DMA descriptor (CDNA5 ISA):
    // 2-D tensor 128x128 of 4-byte elems, full-tile load, LDS padding of
    // 4 DWORDs after every 128 DWORDs => LDS row stride 132 floats.
    const unsigned long long ga = (unsigned long long)(size_t)A_g;
    const unsigned lds_base = (unsigned)(size_t)A_s;   // flat LDS addr[31:0]

    u32x4 g0;
    g0.x = 1u;                                         // count=1 (valid user D#)
    g0.y = lds_base;                                   // lds_addr [63:32]
    g0.z = (unsigned)(ga & 0xFFFFFFFFull);             // global_addr lo
    g0.w = (unsigned)((ga >> 32) & 0x01FFFFFFull)      // global_addr [56:32]
         | (2u << 30);                                 // type=2 ("image")

    i32x8 g1;
    g1[0] = (int)((2u << 16)     // data_size = 2 -> 4-byte elements
                | (1u << 20)     // pad_enable
                | (6u << 22)     // pad_interval: 128 DWORDs
                | (3u << 25));   // pad_amount:   4 DWORDs
    g1[1] = (int)(128u << 16);   // tensor_dim0[15:0] at bits 63:48
    g1[2] = (int)(128u << 16);   // tensor_dim1[15:0] at bits 95:80
    g1[3] = (int)(128u << 16);   // tile_dim0 = 128 at bits 127:112
    g1[4] = (int)(128u);         // tile_dim1 = 128 at bits 143:128
    g1[5] = (int)(128u);         // tensor_dim0_stride lo32 = 128 elems
    g1[6] = (int)(16384u << 16); // tensor_dim1_stride lo16 at bits 223:208
    g1[7] = 0;

    i32x4 gz  = {0, 0, 0, 0};                  // groups 2/3 unused (2-D tile)
    i32x8 gz8 = {0, 0, 0, 0, 0, 0, 0, 0};      // extra arg of 6-arg builtin

    // clang-23 / therock-10.0 form: 6 args, 5th is int32x8 (zero-filled ok)
    __builtin_amdgcn_tensor_load_to_lds(g0, g1, gz, gz, gz8, 0);
#if __has_builtin(__builtin_amdgcn_s_wait_tensorcnt)
    __builtin_amdgcn_s_wait_tensorcnt(0);
#else
    asm volatile("s_wait_tensorcnt 0x0" ::: "memory");
#endif
  }
#else
  for (int r = wave; r < N_ASSETS; r += 8) {
    float4 val = reinterpret_cast<const float4*>(A_g + (size_t)r * N_ASSETS)[lane];
    reinterpret_cast<float4*>(A_s + r * LSTRIDE)[lane] = val;
  }
#endif
  if (tid < N_ASSETS) {
    rets_s[tid] = rets[(size_t)b * N_ASSETS + tid];
    w_s[tid]    = 1.0f / N_ASSETS;
  }
  __syncthreads();

  // ------------------------------------------------------------------
  // Phase 1: Q = gamma^2 * A^T A via V_WMMA_F32_16X16X4_F32.
  // Wave w builds Q rows [16w, 16w+16) x all 128 cols (8 tiles), K in 32x4.
  // fp32 A-operand layout (ISA 7.12.2): lanes 0-15 -> M, half-wave selects
  // K pair {0,1} vs {2,3}; VGPR v -> K = base + v. B-operand symmetric in N.
  // ------------------------------------------------------------------
  const float g0v = gamma[b];
  const float g2 = g0v * g0v;

  v8f acc[8] = {};
  const int l15   = lane & 15;
  const int kbase = (lane >> 4) << 1;          // 0 for lanes 0-15, 2 for 16-31
  const int colA  = wave * 16 + l15;

  for (int it = 0; it < 32; ++it) {
    const int row = it * 4 + kbase;
    v2f a;
    a.x = A_s[row       * LSTRIDE + colA];
    a.y = A_s[(row + 1) * LSTRIDE + colA];
#pragma unroll
    for (int kt = 0; kt < 8; ++kt) {
      const int colB = kt * 16 + l15;
      v2f bb;
      bb.x = A_s[row       * LSTRIDE + colB];
      bb.y = A_s[(row + 1) * LSTRIDE + colB];
      acc[kt] = __builtin_amdgcn_wmma_f32_16x16x4_f32(
          /*neg_a=*/false, a, /*neg_b=*/false, bb,
          /*c_mod=*/(short)0, acc[kt], /*reuse_a=*/false, /*reuse_b=*/false);
    }
  }

  // Scale by gamma^2, spill tiles to Q_s, and accumulate ||Q||_F^2.
  // C/D layout: VGPR r -> row M = r (lanes 0-15) / 8+r (lanes 16-31), N = lane&15.
  float ss = 0.0f;
  const int mbase = wave * 16 + ((lane >> 4) << 3);
#pragma unroll
  for (int kt = 0; kt < 8; ++kt) {
#pragma unroll
    for (int r = 0; r < 8; ++r) {
      float q = acc[kt][r] * g2;
      ss += q * q;
      Q_s[(mbase + r) * LSTRIDE + kt * 16 + l15] = q;
    }
  }
  ss = wred_sum(ss);
  if (lane == 0) red_s[wave] = ss;
  __syncthreads();
  if (tid == 0) {
    float S = 0.0f;
#pragma unroll
    for (int i = 0; i < 8; ++i) S += red_s[i];
    step_s[0] = 1.0f / (2.0f * sqrtf(S) + 1e-6f);
  }
  __syncthreads();
  const float step = step_s[0];

  // ------------------------------------------------------------------
  // Phase 2: 500 PGD iterations, everything LDS-resident.
  //   matvec: 2 threads per row (64 fp32 FMAs each, float4 LDS reads)
  //   projection: redundant per-wave bisection (no cross-wave traffic)
  // ------------------------------------------------------------------
  const int   i_row  = tid >> 1;
  const int   hh     = tid & 1;
  const float4* Qrow = reinterpret_cast<const float4*>(Q_s + i_row * LSTRIDE + hh * 64);
  const float4* whal = reinterpret_cast<const float4*>(w_s + hh * 64);
  const float my_ret = rets_s[i_row];

  for (int iter = 0; iter < PGD_ITERS; ++iter) {
    // --- gradient step: v = w - step * (2 Q w - rets) ---
    float p = 0.0f;
#pragma unroll
    for (int j = 0; j < 16; ++j) {
      float4 q4 = Qrow[j];
      float4 w4 = whal[j];
      p += q4.x * w4.x + q4.y * w4.y + q4.z * w4.z + q4.w * w4.w;
    }
    float tot = p + __shfl_xor(p, 1, 32);
    float g   = 2.0f * tot - my_ret;
    float v   = w_s[i_row] - step * g;
    if (hh == 0) v_s[i_row] = v;
    __syncthreads();

    // --- capped-simplex projection (every wave redundantly) ---
    float vc0 = v_s[lane];
    float vc1 = v_s[lane + 32];
    float vc2 = v_s[lane + 64];
    float vc3 = v_s[lane + 96];

    float lo = wred_min(fminf(fminf(vc0, vc1), fminf(vc2, vc3))) - MW;
    float hi = wred_max(fmaxf(fmaxf(vc0, vc1), fmaxf(vc2, vc3)));

    for (int bi = 0; bi < BIS_ITERS; ++bi) {
      float mid = 0.5f * (lo + hi);
      float s = clip01(vc0 - mid) + clip01(vc1 - mid)
              + clip01(vc2 - mid) + clip01(vc3 - mid);
      s = wred_sum(s);
      bool gt = s > 1.0f;
      lo = gt ? mid : lo;
      hi = gt ? hi : mid;
    }
    float tau0 = 0.5f * (lo + hi);

    // exact refinement on the identified active set
    float sf = 0.0f, su = 0.0f, sv = 0.0f;
    {
      float d;
      d = vc0 - tau0; { bool fr = (d > 0.0f) & (d < MW); bool up = (d >= MW);
                        sf += fr ? 1.0f : 0.0f; su += up ? 1.0f : 0.0f; sv += fr ? vc0 : 0.0f; }
      d = vc1 - tau0; { bool fr = (d > 0.0f) & (d < MW); bool up = (d >= MW);
                        sf += fr ? 1.0f : 0.0f; su += up ? 1.0f : 0.0f; sv += fr ? vc1 : 0.0f; }
      d = vc2 - tau0; { bool fr = (d > 0.0f) & (d < MW); bool up = (d >= MW);
                        sf += fr ? 1.0f : 0.0f; su += up ? 1.0f : 0.0f; sv += fr ? vc2 : 0.0f; }
      d = vc3 - tau0; { bool fr = (d > 0.0f) & (d < MW); bool up = (d >= MW);
                        sf += fr ? 1.0f : 0.0f; su += up ? 1.0f : 0.0f; sv += fr ? vc3 : 0.0f; }
    }
    sf = wred_sum(sf);
    su = wred_sum(su);
    sv = wred_sum(sv);
    float k   = fmaxf(sf, 1.0f);
    float tau = (sv + MW * su - 1.0f) / k;

    float w0 = clip01(vc0 - tau);
    float w1 = clip01(vc1 - tau);
    float w2 = clip01(vc2 - tau);
    float w3 = clip01(vc3 - tau);

    // waves 0..3 each own one 32-wide chunk of w
    if (wave < 4) {
      float wv = (wave == 0) ? w0 : (wave == 1) ? w1 : (wave == 2) ? w2 : w3;
      w_s[lane + 32 * wave] = wv;
    }
    __syncthreads();
  }

  if (tid < N_ASSETS) out[(size_t)b * N_ASSETS + tid] = w_s[tid];
}

extern "C" void kernel_launch(void* const* d_in, const int* in_sizes, int n_in,
                              void* d_out, int out_size, void* d_ws, size_t ws_size,
                              hipStream_t stream) {
  const float* rets  = (const float*)d_in[0];
  const float* cov   = (const float*)d_in[1];
  const float* gamma = (const float*)d_in[2];
  float* out = (float*)d_out;

  const int n_samples = in_sizes[0] / N_ASSETS;   // 2048
  const size_t smem_bytes =
      (size_t)(2 * N_ASSETS * LSTRIDE + 3 * N_ASSETS + 16) * sizeof(float);

  PortfolioPGD_kernel<<<n_samples, 256, smem_bytes, stream>>>(rets, cov, gamma, out);
}